// COAT_block_33526514712806
// MI455X (gfx1250) — compile-verified
//
#include <hip/hip_runtime.h>
#include <hip/hip_bf16.h>

typedef float v2f __attribute__((ext_vector_type(2)));
typedef float v8f __attribute__((ext_vector_type(8)));

#define BB     4
#define NNODE  10000
#define TT     12
#define BNN    40000          // BB*NNODE
#define EE     160000
#define CFc    64
#define FTc    64
#define NHEAD  4
#define MROWS  480000         // BNN*TT
#define OUTHALF 30720000      // BB*NNODE*FTc*TT

// ---------------- CDNA5 async global->LDS staging ----------------
__device__ __forceinline__ void async_g2l(const void* gbase, int byteoff, void* ldsdst) {
  unsigned lds = (unsigned)(unsigned long long)ldsdst;   // low 32 bits of generic addr == LDS offset
  asm volatile("global_load_async_to_lds_b32 %0, %1, %2"
               :: "v"(lds), "v"(byteoff), "s"(gbase) : "memory");
}
__device__ __forceinline__ void async_wait() {
  asm volatile("s_wait_asynccnt 0" ::: "memory");
}

// ---------------- small utility kernels ----------------
__global__ void k_fill(float* __restrict__ p, float v, size_t n) {
  size_t gid = (size_t)blockIdx.x * blockDim.x + threadIdx.x;
  if (gid < n) p[gid] = v;
}

__global__ void k_deg(const int* __restrict__ ei, float* __restrict__ deg) {
  int e = blockIdx.x * blockDim.x + threadIdx.x;
  if (e < EE) atomicAdd(&deg[ei[e]], 1.0f);
}

__global__ void k_dis(float* __restrict__ dis) {
  int i = blockIdx.x * blockDim.x + threadIdx.x;
  if (i < BNN) { float d = dis[i]; dis[i] = (d > 0.f) ? rsqrtf(d) : 0.f; }
}

__global__ void k_we(const int* __restrict__ ei, const float* __restrict__ dis,
                     float* __restrict__ w_e) {
  int e = blockIdx.x * blockDim.x + threadIdx.x;
  if (e < EE) w_e[e] = -dis[ei[e]] * dis[ei[EE + e]];
}

__global__ void k_xt(const float* __restrict__ x, float* __restrict__ xt) {
  int gid = blockIdx.x * blockDim.x + threadIdx.x;   // bn*12 + t
  if (gid < MROWS) {
    int bn = gid / TT, t = gid - bn * TT;
    xt[t * BNN + bn] = x[gid];
  }
}

__global__ void k_prop(const int* __restrict__ ei, const float* __restrict__ w_e,
                       const float* __restrict__ vin, float* __restrict__ vout) {
  int gid = blockIdx.x * blockDim.x + threadIdx.x;   // e + t*EE
  if (gid < EE * TT) {
    int e = gid % EE, t = gid / EE;
    int s = ei[e], d = ei[EE + e];
    atomicAdd(&vout[t * BNN + d], w_e[e] * vin[t * BNN + s]);
  }
}

__global__ void k_txfin(float* __restrict__ tx2, const float* __restrict__ xt) {
  int i = blockIdx.x * blockDim.x + threadIdx.x;
  if (i < MROWS) tx2[i] = 2.0f * tx2[i] - xt[i];
}

__global__ void k_h(const float* __restrict__ xt, const float* __restrict__ tx1,
                    const float* __restrict__ tx2, const float* __restrict__ cw,
                    const float* __restrict__ cb, float* __restrict__ h) {
  long long gid = (long long)blockIdx.x * blockDim.x + threadIdx.x;
  if (gid < (long long)MROWS * 64) {
    int cf = (int)(gid & 63); long long row = gid >> 6;
    h[gid] = cw[cf] * xt[row] + cw[64 + cf] * tx1[row] + cw[128 + cf] * tx2[row] + cb[cf];
  }
}

// ---------------- WMMA fp32 GEMM: C[M,ND] = A[M,KD] @ W[ND,KD]^T (+bias) ----------------
// LDS weight layout is pair-interleaved so each B fragment is one aligned ds_load_b64:
//   Wt[(k>>1)*(2*ND) + 2*n + (k&1)] = W[n*KD + k]
template<int KD, int ND, bool HASB>
__global__ void gemm_wmma(const float* __restrict__ A, const float* __restrict__ W,
                          const float* __restrict__ bias, float* __restrict__ Out, int M) {
  constexpr int NT = ND / 16;
  __shared__ __align__(16) float Wt[KD * ND];
  for (int idx = threadIdx.x; idx < KD * ND; idx += blockDim.x) {
    int kk2 = idx / (2 * ND); int rem = idx - kk2 * (2 * ND);
    int n = rem >> 1; int k = kk2 * 2 + (rem & 1);
    async_g2l(W, (n * KD + k) * 4, &Wt[idx]);
  }
  async_wait();
  __syncthreads();

  int wave = threadIdx.x >> 5, lane = threadIdx.x & 31;
  int mtile = blockIdx.x * (blockDim.x >> 5) + wave;
  if (mtile * 16 >= M) return;
  const float* Arow = A + (size_t)mtile * 16 * KD;
  __builtin_prefetch(Arow, 0, 1);
  int mrow = lane & 15;
  int khalf = (lane >> 4) << 1;                        // lanes 16..31 hold K+2,K+3

  v8f acc[NT];
  v8f vz = {0.f,0.f,0.f,0.f,0.f,0.f,0.f,0.f};
#pragma unroll
  for (int nt = 0; nt < NT; ++nt) acc[nt] = vz;

  for (int kb = 0; kb < KD; kb += 4) {
    int k0 = kb + khalf;                               // even
    v2f a = *(const v2f*)(Arow + mrow * KD + k0);
    const float* bp = Wt + (k0 >> 1) * (2 * ND);
#pragma unroll
    for (int nt = 0; nt < NT; ++nt) {
      v2f b = *(const v2f*)(bp + 2 * (nt * 16 + mrow));
      acc[nt] = __builtin_amdgcn_wmma_f32_16x16x4_f32(false, a, false, b,
                                                      (short)0, acc[nt], false, false);
    }
  }
  int half = lane >> 4;
#pragma unroll
  for (int nt = 0; nt < NT; ++nt)
#pragma unroll
    for (int r = 0; r < 8; ++r) {
      int row = mtile * 16 + r + half * 8;
      int col = nt * 16 + mrow;
      float v = acc[nt][r];
      if (HASB) v += bias[col];
      Out[(size_t)row * ND + col] = v;
    }
}

// ---------------- WMMA fp32 1x3 conv along T as im2col GEMM (K=192) ----------------
__device__ __forceinline__ float conv_fetchA(const float* __restrict__ In, int bn, int tt, int k) {
  int cf = k / 3, dt = k - cf * 3;
  int ti = tt + dt - 1;
  return (ti >= 0 && ti < TT) ? In[((size_t)bn * TT + ti) * 64 + cf] : 0.f;
}

template<bool FINAL>
__global__ void conv3_wmma(const float* __restrict__ In, const float* __restrict__ W,
                           const float* __restrict__ bias, float* __restrict__ Out) {
  constexpr int KD = 192, ND = 64, NT = 4;
  __shared__ __align__(16) float Wt[KD * ND];          // 48 KB, pair-interleaved
  for (int idx = threadIdx.x; idx < KD * ND; idx += blockDim.x) {
    int kk2 = idx / (2 * ND); int rem = idx - kk2 * (2 * ND);
    int n = rem >> 1; int k = kk2 * 2 + (rem & 1);
    async_g2l(W, (n * KD + k) * 4, &Wt[idx]);
  }
  async_wait();
  __syncthreads();

  int wave = threadIdx.x >> 5, lane = threadIdx.x & 31;
  int mtile = blockIdx.x * (blockDim.x >> 5) + wave;
  if (mtile * 16 >= MROWS) return;
  int mrow = lane & 15;
  int khalf = (lane >> 4) << 1;
  int row_g = mtile * 16 + mrow;
  int bn = row_g / TT, tt = row_g - bn * TT;
  __builtin_prefetch(In + (size_t)bn * TT * 64, 0, 1);

  v8f acc[NT];
  v8f vz = {0.f,0.f,0.f,0.f,0.f,0.f,0.f,0.f};
#pragma unroll
  for (int nt = 0; nt < NT; ++nt) acc[nt] = vz;

  for (int kb = 0; kb < KD; kb += 4) {
    int k0 = kb + khalf;                               // even
    v2f a;
    a.x = conv_fetchA(In, bn, tt, k0);
    a.y = conv_fetchA(In, bn, tt, k0 + 1);
    const float* bp = Wt + (k0 >> 1) * (2 * ND);
#pragma unroll
    for (int nt = 0; nt < NT; ++nt) {
      v2f b = *(const v2f*)(bp + 2 * (nt * 16 + mrow));
      acc[nt] = __builtin_amdgcn_wmma_f32_16x16x4_f32(false, a, false, b,
                                                      (short)0, acc[nt], false, false);
    }
  }
  int half = lane >> 4;
#pragma unroll
  for (int nt = 0; nt < NT; ++nt)
#pragma unroll
    for (int r = 0; r < 8; ++r) {
      int row = mtile * 16 + r + half * 8;
      int col = nt * 16 + mrow;
      float v = acc[nt][r] + bias[col];
      if (FINAL) {
        int bnr = row / TT, tr = row - bnr * TT;       // (B,N,FT,T) layout
        Out[((size_t)bnr * 64 + col) * TT + tr] = v;
      } else {
        Out[(size_t)row * 64 + col] = v;
      }
    }
}

// ---------------- GAT attention pieces ----------------
__global__ void k_asd(const float* __restrict__ xp, const float* __restrict__ as_w,
                      const float* __restrict__ ad_w, float* __restrict__ a_s,
                      float* __restrict__ a_d) {
  int gid = blockIdx.x * blockDim.x + threadIdx.x;     // row*2 + h
  if (gid < MROWS * 2) {
    int h = gid & 1; size_t row = (size_t)(gid >> 1);
    float s = 0.f, d = 0.f;
    for (int cf = 0; cf < 64; ++cf) {
      float v = xp[row * 128 + h * 64 + cf];
      s += v * as_w[h * 64 + cf];
      d += v * ad_w[h * 64 + cf];
    }
    a_s[row * 2 + h] = s; a_d[row * 2 + h] = d;
  }
}

__device__ __forceinline__ void edge_sd(const int* __restrict__ ei, int i, int& s, int& d) {
  if (i < EE) { s = ei[i]; d = ei[EE + i]; } else { s = i - EE; d = s; }
}

__device__ __forceinline__ void atomicMaxF(float* a, float v) {
  unsigned int* ai = (unsigned int*)a;
  unsigned int old = *ai, assumed;
  do {
    assumed = old;
    if (__uint_as_float(assumed) >= v) break;
    old = atomicCAS(ai, assumed, __float_as_uint(v));
  } while (old != assumed);
}

__global__ void k_emax(const int* __restrict__ ei, const float* __restrict__ a_s,
                       const float* __restrict__ a_d, float* __restrict__ mx) {
  int gid = blockIdx.x * blockDim.x + threadIdx.x;
  if (gid < (EE + BNN) * TT) {
    int i = gid % (EE + BNN), t = gid / (EE + BNN);
    int s, d; edge_sd(ei, i, s, d);
    size_t rs = (size_t)t * BNN + s, rd = (size_t)t * BNN + d;
    for (int h = 0; h < 2; ++h) {
      float e = a_s[rs * 2 + h] + a_d[rd * 2 + h];
      e = (e > 0.f) ? e : 0.2f * e;
      atomicMaxF(&mx[rd * 2 + h], e);
    }
  }
}

__global__ void k_esum(const int* __restrict__ ei, const float* __restrict__ a_s,
                       const float* __restrict__ a_d, const float* __restrict__ mx,
                       float* __restrict__ zz) {
  int gid = blockIdx.x * blockDim.x + threadIdx.x;
  if (gid < (EE + BNN) * TT) {
    int i = gid % (EE + BNN), t = gid / (EE + BNN);
    int s, d; edge_sd(ei, i, s, d);
    size_t rs = (size_t)t * BNN + s, rd = (size_t)t * BNN + d;
    for (int h = 0; h < 2; ++h) {
      float e = a_s[rs * 2 + h] + a_d[rd * 2 + h];
      e = (e > 0.f) ? e : 0.2f * e;
      atomicAdd(&zz[rd * 2 + h], expf(e - mx[rd * 2 + h]));
    }
  }
}

__global__ void k_eagg(const int* __restrict__ ei, const float* __restrict__ a_s,
                       const float* __restrict__ a_d, const float* __restrict__ mx,
                       const float* __restrict__ zz, const float* __restrict__ xp,
                       float* __restrict__ agg) {
  long long gid = (long long)blockIdx.x * blockDim.x + threadIdx.x;
  if (gid < (long long)(EE + BNN) * TT * 64) {
    int cf = (int)(gid & 63);
    long long j = gid >> 6;
    int i = (int)(j % (EE + BNN)), t = (int)(j / (EE + BNN));
    int s, d; edge_sd(ei, i, s, d);
    size_t rs = (size_t)t * BNN + s, rd = (size_t)t * BNN + d;
    for (int h = 0; h < 2; ++h) {
      float e = a_s[rs * 2 + h] + a_d[rd * 2 + h];
      e = (e > 0.f) ? e : 0.2f * e;
      float alpha = expf(e - mx[rd * 2 + h]) / zz[rd * 2 + h];
      atomicAdd(&agg[rd * 128 + h * 64 + cf], alpha * xp[rs * 128 + h * 64 + cf]);
    }
  }
}

__global__ void k_mean(const float* __restrict__ agg, const float* __restrict__ gb,
                       float* __restrict__ spat) {
  long long gid = (long long)blockIdx.x * blockDim.x + threadIdx.x;
  if (gid < (long long)MROWS * 64) {
    int cf = (int)(gid & 63);
    long long r = gid >> 6;                  // bn*12 + t
    int bn = (int)(r / TT), t = (int)(r - (long long)bn * TT);
    size_t rt = (size_t)t * BNN + bn;
    spat[r * 64 + cf] = 0.5f * (agg[rt * 128 + cf] + agg[rt * 128 + 64 + cf]) + gb[cf];
  }
}

// ---------------- per-node MHA (T=12, 4 heads, dh=16) ----------------
__global__ void k_attn(const float* __restrict__ qkv, float* __restrict__ O) {
  int gid = blockIdx.x * blockDim.x + threadIdx.x;     // bn*(4*12) + h*12 + tq
  if (gid >= BNN * NHEAD * TT) return;
  int tq = gid % TT; int j = gid / TT;
  int h = j % NHEAD; int bn = j / NHEAD;
  const float* base = qkv + (size_t)bn * TT * 192;
  float q[16];
#pragma unroll
  for (int d = 0; d < 16; ++d) q[d] = base[tq * 192 + h * 16 + d];
  float sc[TT]; float m = -1e30f;
#pragma unroll
  for (int tk = 0; tk < TT; ++tk) {
    float s = 0.f;
#pragma unroll
    for (int d = 0; d < 16; ++d) s += q[d] * base[tk * 192 + 64 + h * 16 + d];
    s *= 0.25f;                                        // 1/sqrt(16)
    sc[tk] = s; m = fmaxf(m, s);
  }
  float z = 0.f;
#pragma unroll
  for (int tk = 0; tk < TT; ++tk) { sc[tk] = expf(sc[tk] - m); z += sc[tk]; }
  float iz = 1.f / z;
#pragma unroll
  for (int d = 0; d < 16; ++d) {
    float o = 0.f;
#pragma unroll
    for (int tk = 0; tk < TT; ++tk) o += sc[tk] * base[tk * 192 + 128 + h * 16 + d];
    O[((size_t)bn * TT + tq) * 64 + h * 16 + d] = o * iz;
  }
}

// ---------------- residual + relu + layernorm(FT) ----------------
__global__ void k_ln(const float* __restrict__ attn, const float* __restrict__ x,
                     const float* __restrict__ rw, const float* __restrict__ rb,
                     const float* __restrict__ g, const float* __restrict__ b,
                     float* __restrict__ out) {
  int r = blockIdx.x * (blockDim.x >> 5) + (threadIdx.x >> 5);   // row = bn*12 + t
  if (r >= MROWS) return;
  int lane = threadIdx.x & 31;
  int bn = r / TT, t = r - bn * TT;
  float xv = x[bn * TT + t];
  float z0 = rw[lane] * xv + rb[lane] + attn[(size_t)r * 64 + lane];
  float z1 = rw[lane + 32] * xv + rb[lane + 32] + attn[(size_t)r * 64 + lane + 32];
  z0 = fmaxf(z0, 0.f); z1 = fmaxf(z1, 0.f);
  float s = z0 + z1;
  for (int off = 16; off > 0; off >>= 1) s += __shfl_xor(s, off, 32);
  float mu = s * (1.f / 64.f);
  float d0 = z0 - mu, d1 = z1 - mu;
  float v = d0 * d0 + d1 * d1;
  for (int off = 16; off > 0; off >>= 1) v += __shfl_xor(v, off, 32);
  float inv = rsqrtf(v * (1.f / 64.f) + 1e-5f);
  out[(size_t)r * 64 + lane]      = d0 * inv * g[lane] + b[lane];
  out[(size_t)r * 64 + lane + 32] = d1 * inv * g[lane + 32] + b[lane + 32];
}

// ---------------- host launcher ----------------
static inline unsigned gblk(long long n) { return (unsigned)((n + 255) / 256); }

extern "C" void kernel_launch(void* const* d_in, const int* in_sizes, int n_in,
                              void* d_out, int out_size, void* d_ws, size_t ws_size,
                              hipStream_t stream) {
  const float* x       = (const float*)d_in[0];
  const int*   ei      = (const int*)  d_in[1];
  const float* cheb_w  = (const float*)d_in[2];
  const float* cheb_b  = (const float*)d_in[3];
  const float* gat_w   = (const float*)d_in[4];
  const float* att_src = (const float*)d_in[5];
  const float* att_dst = (const float*)d_in[6];
  const float* gat_b   = (const float*)d_in[7];
  const float* tc_w    = (const float*)d_in[8];
  const float* tc_b    = (const float*)d_in[9];
  const float* in_w    = (const float*)d_in[10];
  const float* in_b    = (const float*)d_in[11];
  const float* out_w   = (const float*)d_in[12];
  const float* out_b   = (const float*)d_in[13];
  const float* res_w   = (const float*)d_in[14];
  const float* res_b   = (const float*)d_in[15];
  const float* ln_g    = (const float*)d_in[16];
  const float* ln_b    = (const float*)d_in[17];
  const float* bc_w    = (const float*)d_in[18];
  const float* bc_b    = (const float*)d_in[19];
  const float* fc_w    = (const float*)d_in[20];
  const float* fc_b    = (const float*)d_in[21];
  float* out = (float*)d_out;

  float* ws = (float*)d_ws;
  size_t off = 0;
  auto alloc = [&](size_t n) { size_t r = off; off += (n + 255) & ~(size_t)255; return ws + r; };
  float* dis = alloc(40000);
  float* w_e = alloc(160000);
  float* xt  = alloc(480000);
  float* tx1 = alloc(480000);
  float* tx2 = alloc(480000);
  float* a_s = alloc(960000);
  float* a_d = alloc(960000);
  float* mx  = alloc(960000);
  float* zz  = alloc(960000);
  float* RA  = alloc(30720000);   // h -> spat -> attn
  float* RB  = alloc(61440000);   // xp -> qkv (qkv spills into RC, which is dead by then)
  float* RC  = alloc(61440000);   // agg
  float* RD  = alloc(30720000);   // tc -> o -> ln_out
  float* qkv = RB;

  // --- Cheb norm & propagation ---
  k_fill<<<gblk(40000), 256, 0, stream>>>(dis, 0.f, 40000);
  k_deg <<<gblk(EE), 256, 0, stream>>>(ei, dis);
  k_dis <<<gblk(BNN), 256, 0, stream>>>(dis);
  k_we  <<<gblk(EE), 256, 0, stream>>>(ei, dis, w_e);
  k_xt  <<<gblk(MROWS), 256, 0, stream>>>(x, xt);
  k_fill<<<gblk(MROWS), 256, 0, stream>>>(tx1, 0.f, MROWS);
  k_prop<<<gblk((long long)EE * TT), 256, 0, stream>>>(ei, w_e, xt, tx1);
  k_fill<<<gblk(MROWS), 256, 0, stream>>>(tx2, 0.f, MROWS);
  k_prop<<<gblk((long long)EE * TT), 256, 0, stream>>>(ei, w_e, tx1, tx2);
  k_txfin<<<gblk(MROWS), 256, 0, stream>>>(tx2, xt);
  k_h   <<<gblk((long long)MROWS * 64), 256, 0, stream>>>(xt, tx1, tx2, cheb_w, cheb_b, RA);

  // --- GAT: projection GEMM (WMMA), attention coefficients, edge softmax, aggregate ---
  gemm_wmma<64, 128, false><<<3750, 256, 0, stream>>>(RA, gat_w, nullptr, RB, MROWS);
  k_asd <<<gblk((long long)MROWS * 2), 256, 0, stream>>>(RB, att_src, att_dst, a_s, a_d);
  k_fill<<<gblk(960000), 256, 0, stream>>>(mx, -1e30f, 960000);
  k_fill<<<gblk(960000), 256, 0, stream>>>(zz, 0.f, 960000);
  k_emax<<<gblk((long long)(EE + BNN) * TT), 256, 0, stream>>>(ei, a_s, a_d, mx);
  k_esum<<<gblk((long long)(EE + BNN) * TT), 256, 0, stream>>>(ei, a_s, a_d, mx, zz);
  k_fill<<<gblk(61440000), 256, 0, stream>>>(RC, 0.f, 61440000);
  k_eagg<<<gblk((long long)(EE + BNN) * TT * 64), 256, 0, stream>>>(ei, a_s, a_d, mx, zz, RB, RC);
  k_mean<<<gblk((long long)MROWS * 64), 256, 0, stream>>>(RC, gat_b, RA);   // spat in RA

  // --- temporal conv (WMMA im2col) -> MHA (qkv GEMM, small attn, out GEMM) ---
  conv3_wmma<false><<<3750, 256, 0, stream>>>(RA, tc_w, tc_b, RD);          // tc in RD
  gemm_wmma<64, 192, true><<<3750, 256, 0, stream>>>(RD, in_w, in_b, qkv, MROWS);
  k_attn<<<gblk((long long)BNN * NHEAD * TT), 256, 0, stream>>>(qkv, RD);   // o in RD
  gemm_wmma<64, 64, true><<<3750, 256, 0, stream>>>(RD, out_w, out_b, RA, MROWS); // attn in RA

  // --- residual + relu + layernorm, then backcast/forecast convs (WMMA) ---
  k_ln<<<(MROWS + 7) / 8, 256, 0, stream>>>(RA, x, res_w, res_b, ln_g, ln_b, RD);
  conv3_wmma<true><<<3750, 256, 0, stream>>>(RD, bc_w, bc_b, out);
  conv3_wmma<true><<<3750, 256, 0, stream>>>(RD, fc_w, fc_b, out + OUTHALF);
}